// KDETorchKNN_41059887350053
// MI455X (gfx1250) — compile-verified
//
#include <hip/hip_runtime.h>
#include <hip/hip_bf16.h>
#include <math.h>

// KDE-kNN on MI455X (gfx1250): fused Gram-tile (V_WMMA_F32_16X16X4_F32) +
// per-row top-K selection on SQUARED distances (sqrt deferred to the end,
// since sqrt(max(.,0)) is monotone). One wave32 per 16-row strip.

typedef __attribute__((ext_vector_type(2))) float v2f;
typedef __attribute__((ext_vector_type(8))) float v8f;

#define M_PTS 8192
#define NGRP  32
#define TILE  16
#define KMAX  16

__global__ __launch_bounds__(32) void kde_zero_hist(int* __restrict__ hist) {
  int t = threadIdx.x;
  if (t < NGRP) hist[t] = 0;
}

__global__ __launch_bounds__(256) void kde_precompute(const float* __restrict__ x,
                                                      const int*   __restrict__ gidx,
                                                      float* __restrict__ sq,
                                                      int*   __restrict__ hist) {
  int i = blockIdx.x * 256 + threadIdx.x;
  if (i < M_PTS) {
    float x0 = x[i * 3 + 0], x1 = x[i * 3 + 1], x2 = x[i * 3 + 2];
    sq[i] = x0 * x0 + x1 * x1 + x2 * x2;
    atomicAdd(&hist[gidx[i]], 1);
  }
}

__global__ __launch_bounds__(32) void kde_knn_kernel(const float* __restrict__ x,
                                                     const int*   __restrict__ gidx,
                                                     const int*   __restrict__ kptr,
                                                     const float* __restrict__ sq,
                                                     const int*   __restrict__ hist,
                                                     float* __restrict__ out) {
  __shared__ float sD[TILE * TILE];        // one 16x16 squared-distance tile
  __shared__ float sCand[TILE * 2 * KMAX]; // 32 candidates per row at the end

  const int   lane = threadIdx.x;
  const int   nIdx = lane & 15;   // column within tile / row within A strip
  const int   hi   = lane >> 4;   // K-half for A/B, M-half for C
  const int   r0   = blockIdx.x * TILE;
  const float FINF = __builtin_inff();

  // ---- A tile (16x4 f32): lanes 0-15 hold K=0,1 ; lanes 16-31 hold K=2,3 ----
  const int arow = r0 + nIdx;
  v2f Amat;
  Amat[0] = hi ? x[arow * 3 + 2] : x[arow * 3 + 0];
  Amat[1] = hi ? 0.0f            : x[arow * 3 + 1];

  // C-tile row metadata for the 8 accumulator rows living in this lane:
  // row(v) = r0 + hi*8 + v  (16x16 f32 C layout)
  float srow[8];
  int   rg[8];
#pragma unroll
  for (int v = 0; v < 8; ++v) {
    int rr  = r0 + hi * 8 + v;
    srow[v] = sq[rr];
    rg[v]   = gidx[rr];
  }

  // Reader role for selection: 2 lanes per row, 8 columns each per tile.
  const int rrow  = lane >> 1;
  const int rhalf = lane & 1;
  float arr[KMAX];                 // ascending sorted top-K list (of d^2)
#pragma unroll
  for (int i = 0; i < KMAX; ++i) arr[i] = FINF;

  for (int c0 = 0; c0 < M_PTS; c0 += TILE) {
    const int col = c0 + nIdx;

    // ---- B tile (4x16 f32): lane%16 = N, lane/16 selects K-half ----
    v2f Bmat;
    Bmat[0] = hi ? x[col * 3 + 2] : x[col * 3 + 0];
    Bmat[1] = hi ? 0.0f           : x[col * 3 + 1];
    const int   gcol = gidx[col];
    const float sqc  = sq[col];

    if (c0 + TILE < M_PTS) {
      // gfx1250 global_prefetch_b8 of next column tile
      __builtin_prefetch(&x[(c0 + TILE + nIdx) * 3], 0, 1);
    }

    // D = A x B : dot(x_row, x_col) for a 16x16 tile (EXEC all-1s here).
    v8f acc = {};
    acc = __builtin_amdgcn_wmma_f32_16x16x4_f32(
        /*neg_a=*/false, Amat, /*neg_b=*/false, Bmat,
        /*c_mod=*/(short)0, acc, /*reuse_a=*/false, /*reuse_b=*/false);

    __syncthreads();  // previous tile fully consumed before overwrite (WAR)
#pragma unroll
    for (int v = 0; v < 8; ++v) {
      // squared distance; sqrt deferred (monotone) to the final reduction
      float d2 = srow[v] + sqc - 2.0f * acc[v];
      if (rg[v] != gcol) d2 = FINF;  // cross-group pairs excluded
      sD[(hi * 8 + v) * TILE + nIdx] = d2;
    }
    __syncthreads();

    // Insert this tile's 8 values for (rrow, rhalf) into the sorted list.
    const float* rp = &sD[rrow * TILE + rhalf * 8];
#pragma unroll
    for (int j = 0; j < 8; ++j) {
      float val = rp[j];
      if (val < arr[KMAX - 1]) {
        arr[KMAX - 1] = val;
#pragma unroll
        for (int t = KMAX - 1; t > 0; --t) {
          float lo = fminf(arr[t - 1], arr[t]);
          float hb = fmaxf(arr[t - 1], arr[t]);
          arr[t - 1] = lo;
          arr[t]     = hb;
        }
      }
    }
  }

  // ---- merge the two half-lists per row and emit density ----
#pragma unroll
  for (int i = 0; i < KMAX; ++i)
    sCand[rrow * (2 * KMAX) + rhalf * KMAX + i] = arr[i];
  __syncthreads();

  if (lane < TILE) {
    int K = kptr[0];
    if (K > KMAX) K = KMAX;
    if (K < 1) K = 1;
    const float* cp = &sCand[lane * (2 * KMAX)];
    float w2 = FINF;
    for (int i = 0; i < 2 * KMAX; ++i) {
      float vi  = cp[i];
      int  rank = 0;
      for (int j = 0; j < 2 * KMAX; ++j) {
        float vj = cp[j];
        rank += (vj < vi) || (vj == vi && j < i);  // tie-broken rank
      }
      if (rank == K - 1) w2 = vi;  // K-th smallest d^2 (self-distance included)
    }
    const int row = r0 + lane;
    const int g   = gidx[row];
    const int c   = hist[g];
    // w = sqrt(max(d2,0)); ni = 3 -> dim = 2 -> vol = pi * w^2 = pi * max(d2,0)
    float wclamped2 = fmaxf(w2, 0.0f);
    float p = (3.14159265358979323846f * wclamped2) / (float)(K - 1);
    if (c < K) p = 1.0f / (float)c;
    out[row] = p;
  }
}

extern "C" void kernel_launch(void* const* d_in, const int* in_sizes, int n_in,
                              void* d_out, int out_size, void* d_ws, size_t ws_size,
                              hipStream_t stream) {
  (void)in_sizes; (void)n_in; (void)out_size; (void)ws_size;

  const float* x    = (const float*)d_in[0];  // [8192, 3] f32
  const int*   gidx = (const int*)d_in[1];    // [8192] i32 group ids
  const int*   kptr = (const int*)d_in[2];    // scalar K (=16)
  float*       out  = (float*)d_out;          // [8192] f32

  float* sq   = (float*)d_ws;                                   // 8192 f32
  int*   hist = (int*)((char*)d_ws + M_PTS * sizeof(float));    // 32 i32

  kde_zero_hist<<<1, 32, 0, stream>>>(hist);
  kde_precompute<<<M_PTS / 256, 256, 0, stream>>>(x, gidx, sq, hist);
  kde_knn_kernel<<<M_PTS / TILE, 32, 0, stream>>>(x, gidx, kptr, sq, hist, out);
}